// SelectiveWKV_1B_61065845015139
// MI455X (gfx1250) — compile-verified
//
#include <hip/hip_runtime.h>

typedef float v2f __attribute__((ext_vector_type(2)));
typedef float v8f __attribute__((ext_vector_type(8)));

static constexpr int D  = 2048;
static constexpr int H  = 32;
static constexpr int HS = 64;
static constexpr int B  = 4;
static constexpr int T  = 2048;
static constexpr int Mrows = B * T;                  // 8192
static constexpr long NBT  = (long)Mrows * D;        // 16,777,216 elements

// Low 32 bits of a generic pointer to LDS == wave-relative LDS byte address
// (ISA 10.2: LDS aperture maps addr[31:0] directly).
__device__ __forceinline__ unsigned lds_off(const void* p) {
  return (unsigned)(unsigned long long)p;
}

// Async global->LDS copies (ASYNCcnt-tracked, CDNA5 ISA).
__device__ __forceinline__ void async_load_b128(unsigned lds, unsigned gofs,
                                                const void* sbase) {
  asm volatile("global_load_async_to_lds_b128 %0, %1, %2 offset:0"
               :: "v"(lds), "v"(gofs), "s"(sbase) : "memory");
}
__device__ __forceinline__ void async_load_b32(unsigned lds, unsigned gofs,
                                               const void* sbase) {
  asm volatile("global_load_async_to_lds_b32 %0, %1, %2 offset:0"
               :: "v"(lds), "v"(gofs), "s"(sbase) : "memory");
}
__device__ __forceinline__ void wait_async0() {
  asm volatile("s_wait_asynccnt 0x0" ::: "memory");
}

// ---------------------------------------------------------------------------
// LayerNorm: one block per (b,t) row. 256 threads, 2048 channels.
// ---------------------------------------------------------------------------
__global__ __launch_bounds__(256)
void ln_kernel(const float* __restrict__ x, const float* __restrict__ g,
               const float* __restrict__ beta, float* __restrict__ xn) {
  const int row = blockIdx.x;
  const int tid = threadIdx.x;
  const float4* x4 = (const float4*)(x + (long)row * D);
  float4* o4 = (float4*)(xn + (long)row * D);
  const float4* g4 = (const float4*)g;
  const float4* b4 = (const float4*)beta;

  float4 v[2];
  float s = 0.f, ss = 0.f;
#pragma unroll
  for (int i = 0; i < 2; ++i) {
    v[i] = x4[tid + i * 256];
    s  += v[i].x + v[i].y + v[i].z + v[i].w;
    ss += v[i].x * v[i].x + v[i].y * v[i].y + v[i].z * v[i].z + v[i].w * v[i].w;
  }

  __shared__ float red[256];
  red[tid] = s; __syncthreads();
  for (int o = 128; o > 0; o >>= 1) { if (tid < o) red[tid] += red[tid + o]; __syncthreads(); }
  const float mu = red[0] * (1.0f / D);
  __syncthreads();
  red[tid] = ss; __syncthreads();
  for (int o = 128; o > 0; o >>= 1) { if (tid < o) red[tid] += red[tid + o]; __syncthreads(); }
  const float var  = red[0] * (1.0f / D) - mu * mu;
  const float rinv = rsqrtf(var + 1e-5f);

#pragma unroll
  for (int i = 0; i < 2; ++i) {
    const float4 gv = g4[tid + i * 256];
    const float4 bv = b4[tid + i * 256];
    float4 ov;
    ov.x = (v[i].x - mu) * rinv * gv.x + bv.x;
    ov.y = (v[i].y - mu) * rinv * gv.y + bv.y;
    ov.z = (v[i].z - mu) * rinv * gv.z + bv.z;
    ov.w = (v[i].w - mu) * rinv * gv.w + bv.w;
    o4[tid + i * 256] = ov;
  }
}

// ---------------------------------------------------------------------------
// GEMM: C[M,N] = A[M,K] @ W[N,K]^T  (torch Linear), fp32 via WMMA 16x16x4.
// Block = 256 threads (8 wave32), C tile 128x128, K panel 32.
// Panels double-buffered in LDS, staged with global_load_async_to_lds_b128.
// EPI: 0 = none, 1 = sigmoid, 2 = +bias then sigmoid.
// ---------------------------------------------------------------------------
template <int EPI>
__global__ __launch_bounds__(256)
void gemm_kernel(const float* __restrict__ A, const float* __restrict__ Wt,
                 const float* __restrict__ bias, float* __restrict__ C) {
  constexpr int Kd  = D;
  constexpr int Nd  = D;
  constexpr int KB  = 32;
  constexpr int LDT = KB + 4;   // 36 floats: 16B-aligned b128 copies, conflict-free reads

  __shared__ float As[2][128 * LDT];
  __shared__ float Bs[2][128 * LDT];

  const int tid  = threadIdx.x;
  const int lane = tid & 31;
  const int wv   = tid >> 5;
  const int lm   = lane & 15;       // N / M index within a 16x16 tile
  const int lh   = lane >> 4;       // K-half (f32 wmma: lanes 16-31 hold K=2,3)
  const int wm0  = (wv >> 2) * 64;  // wave rows: 2 row-groups of 64
  const int wn0  = (wv & 3) * 32;   // wave cols: 4 col-groups of 32

  const long rowBase = (long)blockIdx.y * 128;
  const long colBase = (long)blockIdx.x * 128;

  // Per-thread staging coordinates: 4 x b128 per matrix per panel.
  const int sr = tid >> 1;                 // 0..127 (row within tile)
  const int sk = (tid & 1) << 4;           // 0 or 16 (k-offset, two b128 pairs)

  auto stage = [&](int bufsel, int k0) {
    const unsigned la = lds_off(&As[bufsel][0]);
    const unsigned lb = lds_off(&Bs[bufsel][0]);
#pragma unroll
    for (int q = 0; q < 2; ++q) {
      const int kq = sk + q * 4 * 2;       // 0,8 or 16,24
      const unsigned lofs = (unsigned)((sr * LDT + kq) * 4);
      const unsigned ga = (unsigned)(((rowBase + sr) * Kd + k0 + kq) * 4);
      async_load_b128(la + lofs, ga, A);
      async_load_b128(la + lofs + 16, ga + 16, A);
      const unsigned gb = (unsigned)(((colBase + sr) * Kd + k0 + kq) * 4);
      async_load_b128(lb + lofs, gb, Wt);
      async_load_b128(lb + lofs + 16, gb + 16, Wt);
    }
  };

  v8f acc[4][2] = {};

  stage(0, 0);
  for (int k0 = 0; k0 < Kd; k0 += KB) {
    const int buf = (k0 >> 5) & 1;
    wait_async0();       // my async writes for this panel have landed
    __syncthreads();     // => every wave's writes landed; prev panel fully read
    if (k0 + KB < Kd) stage(buf ^ 1, k0 + KB);

    const float* Asb = As[buf];
    const float* Bsb = Bs[buf];
#pragma unroll
    for (int k4 = 0; k4 < KB; k4 += 4) {
      const int ko = k4 + lh * 2;
      v2f a[4], b[2];
#pragma unroll
      for (int mt = 0; mt < 4; ++mt) {
        const float* p = &Asb[(wm0 + mt * 16 + lm) * LDT + ko];
        a[mt][0] = p[0]; a[mt][1] = p[1];
      }
#pragma unroll
      for (int nt = 0; nt < 2; ++nt) {
        const float* p = &Bsb[(wn0 + nt * 16 + lm) * LDT + ko];
        b[nt][0] = p[0]; b[nt][1] = p[1];
      }
#pragma unroll
      for (int mt = 0; mt < 4; ++mt)
#pragma unroll
        for (int nt = 0; nt < 2; ++nt)
          acc[mt][nt] = __builtin_amdgcn_wmma_f32_16x16x4_f32(
              false, a[mt], false, b[nt], (short)0, acc[mt][nt], false, false);
    }
  }

  // Store per ISA C-layout: VGPR r holds M=r (lanes 0-15) / M=r+8 (lanes 16-31).
#pragma unroll
  for (int mt = 0; mt < 4; ++mt)
#pragma unroll
    for (int nt = 0; nt < 2; ++nt)
#pragma unroll
      for (int r = 0; r < 8; ++r) {
        const long row = rowBase + wm0 + mt * 16 + lh * 8 + r;
        const long col = colBase + wn0 + nt * 16 + lm;
        float val = acc[mt][nt][r];
        if (EPI == 2) val += bias[col];
        if (EPI >= 1) val = 1.0f / (1.0f + __expf(-val));
        C[row * (long)Nd + col] = val;
      }
}

// ---------------------------------------------------------------------------
// Recurrent scan. One block (64 threads) per (b,h); thread j owns S[:,j].
// k/w/r double-buffered in LDS via async copies; next step's operands are
// in flight while the current step's 64-deep FMA chain executes.
// ---------------------------------------------------------------------------
__global__ __launch_bounds__(64)
void scan_kernel(const float* __restrict__ Kp, const float* __restrict__ Vp,
                 const float* __restrict__ Wp, const float* __restrict__ Rp,
                 const float* __restrict__ state_in,
                 float* __restrict__ outs, float* __restrict__ state_out) {
  const int bh = blockIdx.x;          // 0..127
  const int b  = bh >> 5;
  const int h  = bh & 31;
  const int j  = threadIdx.x;         // 0..63

  float S[HS];
  const long sbase = (long)bh * HS * HS;
#pragma unroll
  for (int i = 0; i < HS; ++i) S[i] = state_in[sbase + (long)i * HS + j];

  __shared__ float4 kb4[2][16], wb4[2][16], rb4[2][16];

  const long base0 = (long)b * T * D + (long)h * HS;

  auto stage = [&](int bufsel, int t) {
    const unsigned gofs = (unsigned)((base0 + (long)t * D + j) * 4);
    async_load_b32(lds_off(&kb4[bufsel][0]) + j * 4, gofs, Kp);
    async_load_b32(lds_off(&wb4[bufsel][0]) + j * 4, gofs, Wp);
    async_load_b32(lds_off(&rb4[bufsel][0]) + j * 4, gofs, Rp);
  };

  stage(0, 0);
  long base = base0;
  for (int t = 0; t < T; ++t, base += D) {
    const int buf = t & 1;
    wait_async0();
    __syncthreads();
    if (t + 1 < T) stage(buf ^ 1, t + 1);

    const float vj = Vp[base + j];
    const float4* k4 = kb4[buf];
    const float4* w4 = wb4[buf];
    const float4* r4 = rb4[buf];
    float out = 0.f;
#pragma unroll
    for (int q = 0; q < 16; ++q) {
      const float4 kv = k4[q], wv = w4[q], rv = r4[q];
      float s;
      s = S[4*q+0] * (1.f - wv.x) + kv.x * vj; S[4*q+0] = s; out += rv.x * s;
      s = S[4*q+1] * (1.f - wv.y) + kv.y * vj; S[4*q+1] = s; out += rv.y * s;
      s = S[4*q+2] * (1.f - wv.z) + kv.z * vj; S[4*q+2] = s; out += rv.z * s;
      s = S[4*q+3] * (1.f - wv.w) + kv.w * vj; S[4*q+3] = s; out += rv.w * s;
    }
    outs[base + j] = out;
  }
#pragma unroll
  for (int i = 0; i < HS; ++i) state_out[sbase + (long)i * HS + j] = S[i];
}

// ---------------------------------------------------------------------------
extern "C" void kernel_launch(void* const* d_in, const int* in_sizes, int n_in,
                              void* d_out, int out_size, void* d_ws, size_t ws_size,
                              hipStream_t stream) {
  (void)in_sizes; (void)n_in; (void)out_size; (void)ws_size;

  const float* x    = (const float*)d_in[0];
  const float* st   = (const float*)d_in[1];
  const float* ln_g = (const float*)d_in[2];
  const float* ln_b = (const float*)d_in[3];
  const float* Wx   = (const float*)d_in[4];
  const float* Ww   = (const float*)d_in[5];
  const float* bw   = (const float*)d_in[6];
  const float* Wk   = (const float*)d_in[7];
  const float* Wv   = (const float*)d_in[8];
  const float* Wr   = (const float*)d_in[9];
  const float* Wo   = (const float*)d_in[10];

  float* y         = (float*)d_out;
  float* state_out = y + NBT;

  float* ws0  = (float*)d_ws;
  float* xn   = ws0;
  float* kb   = ws0 + 1 * NBT;
  float* vb   = ws0 + 2 * NBT;
  float* rb   = ws0 + 3 * NBT;
  float* xw   = ws0 + 4 * NBT;
  float* wb   = ws0 + 5 * NBT;
  float* outs = xn;  // xn dead after the projection GEMMs

  ln_kernel<<<Mrows, 256, 0, stream>>>(x, ln_g, ln_b, xn);

  dim3 g(D / 128, Mrows / 128);  // (16, 64)
  gemm_kernel<0><<<g, 256, 0, stream>>>(xn, Wk, nullptr, kb);
  gemm_kernel<0><<<g, 256, 0, stream>>>(xn, Wv, nullptr, vb);
  gemm_kernel<1><<<g, 256, 0, stream>>>(xn, Wr, nullptr, rb);   // r = sigmoid(.)
  gemm_kernel<0><<<g, 256, 0, stream>>>(xn, Wx, nullptr, xw);
  gemm_kernel<2><<<g, 256, 0, stream>>>(xw, Ww, bw, wb);        // w = sigmoid(. + bw)

  scan_kernel<<<B * H, 64, 0, stream>>>(kb, vb, wb, rb, st, outs, state_out);

  gemm_kernel<0><<<g, 256, 0, stream>>>(outs, Wo, nullptr, y);
}